// GNLoss_29540785062233
// MI455X (gfx1250) — compile-verified
//
#include <hip/hip_runtime.h>
#include <hip/hip_bf16.h>

#define BDIM 128
#define WAVES 4
#define SPW 4            // samples per wave
#define SPB 16           // samples per block
#define C_ 128
#define H_ 240
#define W_ 320
#define N_ 1024
#define EPS_ 1e-9f

typedef float v2f __attribute__((ext_vector_type(2)));
typedef float v8f __attribute__((ext_vector_type(8)));

__device__ __forceinline__ float bval(const float* __restrict__ p, int y, int x) {
    return p[y * W_ + x];
}

// One block = 16 samples. Phase 1: gather + bilinear (fused gradient). Phase 2:
// WMMA Gram D = V*V^T per wave (4 samples' 4x4 Grams on the block diagonal).
// Phase 3: per-sample 2x2 GN solve, block partial sums.
__global__ __launch_bounds__(BDIM) void gnloss_main(
    const float* __restrict__ Fa, const float* __restrict__ Fb,
    const float* __restrict__ ma, const float* __restrict__ mb,
    const float* __restrict__ noise, float* __restrict__ partials)
{
    __shared__ float Vlds[WAVES * 16 * C_];   // rows: [s*4 + {Jx,Jy,fs,ft}][c]
    __shared__ float Dlds[WAVES * 256];       // 16x16 Gram result per wave
    __shared__ float red1[SPB], red2[SPB];

    const int tid  = threadIdx.x;
    const int wave = tid >> 5;
    const int lane = tid & 31;

    // ---------------- Phase 1: gather + fused gradient bilinear ----------------
    {
        const int sl   = lane >> 3;    // sample within wave (0..3)
        const int subc = lane & 7;     // channel sublane   (0..7)
        const int sg   = blockIdx.x * SPB + wave * SPW + sl;
        const int b    = sg >> 10;     // N_ == 1024
        const int n    = sg & (N_ - 1);
        const int ci   = (b * N_ + n) * 2;
        const float inv = 0.125f;      // 1/LEVEL

        float ax  = ma[ci] * inv,  ay  = ma[ci + 1] * inv;
        float ubx = mb[ci] * inv,  uby = mb[ci + 1] * inv;
        float xsx = 2.f * noise[ci]     - 1.f + ubx;
        float xsy = 2.f * noise[ci + 1] - 1.f + uby;

        // F_a bilinear setup
        float afx = floorf(ax), afy = floorf(ay);
        float awx = ax - afx,   awy = ay - afy;
        int ax0 = min(max((int)afx, 0), W_ - 1), ax1 = min(ax0 + 1, W_ - 1);
        int ay0 = min(max((int)afy, 0), H_ - 1), ay1 = min(ay0 + 1, H_ - 1);

        // F_b bilinear setup at xs
        float bfx = floorf(xsx), bfy = floorf(xsy);
        float wx = xsx - bfx,    wy = xsy - bfy;
        int x0 = min(max((int)bfx, 0), W_ - 1), x1 = min(x0 + 1, W_ - 1);
        int y0 = min(max((int)bfy, 0), H_ - 1), y1 = min(y0 + 1, H_ - 1);
        // per-corner clamped neighbor indices for central differences
        int x0m = max(x0 - 1, 0), x0p = min(x0 + 1, W_ - 1);
        int x1m = max(x1 - 1, 0), x1p = min(x1 + 1, W_ - 1);
        int y0m = max(y0 - 1, 0), y0p = min(y0 + 1, H_ - 1);
        int y1m = max(y1 - 1, 0), y1p = min(y1 + 1, H_ - 1);

        float w00 = (1.f - wx) * (1.f - wy), w01 = wx * (1.f - wy);
        float w10 = (1.f - wx) * wy,         w11 = wx * wy;
        float q00 = (1.f - awx) * (1.f - awy), q01 = awx * (1.f - awy);
        float q10 = (1.f - awx) * awy,         q11 = awx * awy;

        float* vrow = &Vlds[wave * 16 * C_ + (sl * 4) * C_];
        for (int t = 0; t < 16; ++t) {
            int c = subc + 8 * t;
            const float* pb = Fb + (size_t)(b * C_ + c) * (H_ * W_);
            const float* pa = Fa + (size_t)(b * C_ + c) * (H_ * W_);
            // f_s
            float fsv = bval(pb, y0, x0) * w00 + bval(pb, y0, x1) * w01
                      + bval(pb, y1, x0) * w10 + bval(pb, y1, x1) * w11;
            // Jx: bilinear of on-the-fly gx corners
            float g00 = 0.5f * (bval(pb, y0, x0p) - bval(pb, y0, x0m));
            float g01 = 0.5f * (bval(pb, y0, x1p) - bval(pb, y0, x1m));
            float g10 = 0.5f * (bval(pb, y1, x0p) - bval(pb, y1, x0m));
            float g11 = 0.5f * (bval(pb, y1, x1p) - bval(pb, y1, x1m));
            float jx = g00 * w00 + g01 * w01 + g10 * w10 + g11 * w11;
            // Jy
            float h00 = 0.5f * (bval(pb, y0p, x0) - bval(pb, y0m, x0));
            float h01 = 0.5f * (bval(pb, y0p, x1) - bval(pb, y0m, x1));
            float h10 = 0.5f * (bval(pb, y1p, x0) - bval(pb, y1m, x0));
            float h11 = 0.5f * (bval(pb, y1p, x1) - bval(pb, y1m, x1));
            float jy = h00 * w00 + h01 * w01 + h10 * w10 + h11 * w11;
            // f_t
            float ftv = bval(pa, ay0, ax0) * q00 + bval(pa, ay0, ax1) * q01
                      + bval(pa, ay1, ax0) * q10 + bval(pa, ay1, ax1) * q11;

            vrow[0 * C_ + c] = jx;
            vrow[1 * C_ + c] = jy;
            vrow[2 * C_ + c] = fsv;
            vrow[3 * C_ + c] = ftv;
        }
    }
    __syncthreads();

    // ---------------- Phase 2: WMMA Gram, D = V * V^T ----------------
    // A (16x4 f32): lane m holds A[m][g+2h] in VGPR g (h = lane>>4).
    // B (4x16): row k=g+2h striped over lanes -> B = A^T has the SAME register
    // image, so the same v2f feeds both operands.
    {
        const int r = lane & 15;
        const int h = lane >> 4;
        const float* vb = &Vlds[wave * 16 * C_ + r * C_];
        v8f acc = {};
        for (int kk = 0; kk < 32; ++kk) {
            v2f a;
            a.x = vb[4 * kk + 2 * h];
            a.y = vb[4 * kk + 2 * h + 1];
            acc = __builtin_amdgcn_wmma_f32_16x16x4_f32(
                false, a, false, a, (short)0, acc, false, false);
        }
        // D layout: VGPR g, lanes 0-15 -> row g, col lane; lanes 16-31 -> row g+8
        float* db = &Dlds[wave * 256];
        #pragma unroll
        for (int g = 0; g < 8; ++g) db[(g + 8 * h) * 16 + r] = acc[g];
    }
    __syncthreads();

    // ---------------- Phase 3: per-sample 2x2 GN solve ----------------
    if (lane < SPW) {
        const int s  = lane;
        const int sg = blockIdx.x * SPB + wave * SPW + s;
        const int b  = sg >> 10;
        const int n  = sg & (N_ - 1);
        const int ci = (b * N_ + n) * 2;
        float nx = noise[ci], ny = noise[ci + 1];

        // sample s's 4x4 Gram: rows/cols 4s..4s+3 of the wave's 16x16 D
        const float* G = &Dlds[wave * 256 + (4 * s) * 16 + 4 * s];
        float G00 = G[0],      G01 = G[1],      G02 = G[2],      G03 = G[3];
        float G11 = G[16 + 1], G12 = G[16 + 2], G13 = G[16 + 3];
        float G22 = G[32 + 2], G33 = G[48 + 3];

        float nrs = fmaxf(sqrtf(G22), 1e-12f);   // ||f_s||
        float nrt = fmaxf(sqrtf(G33), 1e-12f);   // ||f_t||
        // b = J^T (f_s/||f_s|| - f_t/||f_t||)
        float b0 = G02 / nrs - G03 / nrt;
        float b1 = G12 / nrs - G13 / nrt;
        // Hm = J^T J + eps I
        float a  = G00 + EPS_, bb = G01, dd = G11 + EPS_;
        float det = a * dd - bb * bb;
        float hb0 = (dd * b0 - bb * b1) / det;   // (Hinv b)_0
        float hb1 = (a  * b1 - bb * b0) / det;   // (Hinv b)_1
        // diff = ub - miu = (ub - xs) + Hinv b, and ub - xs = 1 - 2*noise
        float d0 = (1.f - 2.f * nx) + hb0;
        float d1 = (1.f - 2.f * ny) + hb1;

        red1[wave * SPW + s] = a * d0 * d0 + 2.f * bb * d0 * d1 + dd * d1 * d1;
        red2[wave * SPW + s] = logf(det);
    }
    __syncthreads();

    if (tid == 0) {
        float s1 = 0.f, s2 = 0.f;
        #pragma unroll
        for (int i = 0; i < SPB; ++i) { s1 += red1[i]; s2 += red2[i]; }
        partials[blockIdx.x]       = s1;
        partials[256 + blockIdx.x] = s2;
    }
}

__global__ __launch_bounds__(256) void gnloss_final(
    const float* __restrict__ partials, float* __restrict__ out)
{
    __shared__ float sh1[256], sh2[256];
    const int t = threadIdx.x;
    sh1[t] = partials[t];
    sh2[t] = partials[256 + t];
    __syncthreads();
    for (int o = 128; o > 0; o >>= 1) {
        if (t < o) { sh1[t] += sh1[t + o]; sh2[t] += sh2[t + o]; }
        __syncthreads();
    }
    if (t == 0) {
        float e1 = 0.5f * sh1[0];
        float e2 = 4096.0f * logf(6.283185307179586f) - 0.5f * sh2[0];
        float e  = e1 + (2.0f / 7.0f) * e2;   // E1_LAMDA*e1 + E2_LAMDA*e2
        out[0] = 0.3f * e;                    // GN_LAMDA * e
        out[1] = e1;
        out[2] = e2;
    }
}

extern "C" void kernel_launch(void* const* d_in, const int* in_sizes, int n_in,
                              void* d_out, int out_size, void* d_ws, size_t ws_size,
                              hipStream_t stream) {
    const float* Fa    = (const float*)d_in[0];
    const float* Fb    = (const float*)d_in[1];
    const float* ma    = (const float*)d_in[2];
    const float* mb    = (const float*)d_in[3];
    const float* noise = (const float*)d_in[4];
    float* partials = (float*)d_ws;           // 512 floats of scratch

    // 4096 samples / 16 per block = 256 blocks, 128 threads (4 wave32) each
    gnloss_main<<<256, BDIM, 0, stream>>>(Fa, Fb, ma, mb, noise, partials);
    gnloss_final<<<1, 256, 0, stream>>>(partials, (float*)d_out);
}